// RPNTrainingProposal_43800076485304
// MI455X (gfx1250) — compile-verified
//
#include <hip/hip_runtime.h>
#include <hip/hip_bf16.h>
#include <stdint.h>

#define N_ANCH  262144
#define N_GT    256
#define NBLK    (N_ANCH / 256)      // 1024 groups of 256 anchors (scan/emit granularity)
#define NBLK_A  (N_ANCH / 512)      // 512 blocks for the main kernel (2 anchors/thread)
#define MAX_POS 128
#define TOTAL   256
#define POS_T   0.7f
#define NEG_T   0.3f

typedef unsigned int u32;
typedef __attribute__((ext_vector_type(4))) u32 u32x4_t;
typedef __attribute__((ext_vector_type(4))) int i32x4_t;
typedef __attribute__((ext_vector_type(8))) int i32x8_t;

#define HAS_TDM (__has_builtin(__builtin_amdgcn_tensor_load_to_lds) && \
                 __has_builtin(__builtin_amdgcn_s_wait_tensorcnt))

__device__ __forceinline__ int imin(int a, int b) { return a < b ? a : b; }

// ---------------------------------------------------------------------------
// Kernel 1: pairwise IoU max/argmax + labels + per-group pos/neg counts.
// GT table (4 KB) is DMA'd into LDS via the Tensor Data Mover once per block.
// 2 anchors per thread: each LDS gt broadcast read feeds two IoU chains.
// ---------------------------------------------------------------------------
__global__ __launch_bounds__(256) void iou_label_kernel(
    const float4* __restrict__ anchors,
    const float*  __restrict__ gtb,
    const int*    __restrict__ p_img_h,
    const int*    __restrict__ p_img_w,
    int* __restrict__ labels,
    int* __restrict__ gtidx,
    int* __restrict__ blkpos,
    int* __restrict__ blkneg)
{
    __shared__ struct {
        float4 gt[N_GT];     // first member -> LDS byte offset 0 (TDM dest)
        float  area[N_GT];
        int    cnt[4];       // {pos_g0, neg_g0, pos_g1, neg_g1}
    } sm;

    const int tid = threadIdx.x;
    if (tid < 4) sm.cnt[tid] = 0;

#if HAS_TDM
    if (tid < 32) {   // one wave issues the DMA (EXEC ignored by TDM)
        unsigned long long ga = (unsigned long long)(const void*)gtb;
        u32x4_t g0;
        g0[0] = 1u;                                   // count=1 valid descriptor
        g0[1] = 0u;                                   // lds_addr = 0 (sm.gt)
        g0[2] = (u32)(ga & 0xFFFFFFFFull);            // global_addr[31:0]
        g0[3] = (u32)((ga >> 32) & 0x01FFFFFFull)     // global_addr[56:32]
              | 0x80000000u;                          // type = 2 ("image")
        i32x8_t g1;
        g1[0] = (int)(2u << 16);                      // data_size=4B; wg_mask=0
        g1[1] = (int)(((u32)(N_GT * 4) & 0xFFFFu) << 16); // tensor_dim0=1024 lo16
        g1[2] = (int)(1u << 16);                      // dim0 hi16=0 | tensor_dim1=1
        g1[3] = (int)((u32)(N_GT * 4) << 16);         // dim1 hi16=0 | tile_dim0=1024
        g1[4] = 1;                                    // tile_dim1=1, tile_dim2=0
        g1[5] = N_GT * 4;                             // tensor_dim0_stride lo32
        g1[6] = (int)(((u32)(N_GT * 4) & 0xFFFFu) << 16); // stride0 hi=0|stride1 lo16
        g1[7] = 0;
        i32x4_t z4 = {0, 0, 0, 0};
#if defined(__clang_major__) && (__clang_major__ >= 23)
        i32x8_t z8 = {0, 0, 0, 0, 0, 0, 0, 0};
        __builtin_amdgcn_tensor_load_to_lds(g0, g1, z4, z4, z8, 0);
#else
        __builtin_amdgcn_tensor_load_to_lds(g0, g1, z4, z4, 0);
#endif
        __builtin_amdgcn_s_wait_tensorcnt(0);
    }
#else
    {   // fallback: cooperative vector load (256 threads x 16 B = 4 KB)
        const float4* g4 = (const float4*)gtb;
        sm.gt[tid] = g4[tid];
    }
#endif
    __syncthreads();
    {
        float4 b = sm.gt[tid];
        sm.area[tid] = (b.z - b.x) * (b.w - b.y);
    }
    __syncthreads();

    const int i0 = blockIdx.x * 512 + tid;   // group 2*blockIdx.x
    const int i1 = i0 + 256;                 // group 2*blockIdx.x + 1
    const float4 a0 = anchors[i0];
    const float4 a1 = anchors[i1];
    const float fh = (float)(*p_img_h);
    const float fw = (float)(*p_img_w);
    const bool in0 = (a0.x >= 0.0f) && (a0.y >= 0.0f) && (a0.z <= fh) && (a0.w <= fw);
    const bool in1 = (a1.x >= 0.0f) && (a1.y >= 0.0f) && (a1.z <= fh) && (a1.w <= fw);
    const float areaA0 = (a0.z - a0.x) * (a0.w - a0.y);
    const float areaA1 = (a1.z - a1.x) * (a1.w - a1.y);

    // Division-free running max/argmax of inter/union, two independent chains.
    float bi0 = 0.0f, bu0 = 1e-8f;
    float bi1 = 0.0f, bu1 = 1e-8f;
    int   bj0 = 0,    bj1 = 0;
#pragma unroll 4
    for (int j = 0; j < N_GT; ++j) {
        const float4 g  = sm.gt[j];
        const float  ab = sm.area[j];

        float ih0 = fminf(a0.z, g.z) - fmaxf(a0.x, g.x);
        float iw0 = fminf(a0.w, g.w) - fmaxf(a0.y, g.y);
        ih0 = fmaxf(ih0, 0.0f); iw0 = fmaxf(iw0, 0.0f);
        const float inter0 = ih0 * iw0;
        const float uni0   = areaA0 + ab - inter0;
        const bool  b0     = (inter0 * bu0) > (bi0 * uni0);
        bi0 = b0 ? inter0 : bi0;
        bu0 = b0 ? uni0   : bu0;
        bj0 = b0 ? j      : bj0;

        float ih1 = fminf(a1.z, g.z) - fmaxf(a1.x, g.x);
        float iw1 = fminf(a1.w, g.w) - fmaxf(a1.y, g.y);
        ih1 = fmaxf(ih1, 0.0f); iw1 = fmaxf(iw1, 0.0f);
        const float inter1 = ih1 * iw1;
        const float uni1   = areaA1 + ab - inter1;
        const bool  b1     = (inter1 * bu1) > (bi1 * uni1);
        bi1 = b1 ? inter1 : bi1;
        bu1 = b1 ? uni1   : bu1;
        bj1 = b1 ? j      : bj1;
    }

    const float u0 = fmaxf(bu0, 1e-8f);
    int lab0 = -1;
    if (bi0 > POS_T * u0) lab0 = 1;
    if (bi0 < NEG_T * u0) lab0 = 0;
    if (!in0) lab0 = -1;

    const float u1 = fmaxf(bu1, 1e-8f);
    int lab1 = -1;
    if (bi1 > POS_T * u1) lab1 = 1;
    if (bi1 < NEG_T * u1) lab1 = 0;
    if (!in1) lab1 = -1;

    labels[i0] = lab0;  gtidx[i0] = bj0;
    labels[i1] = lab1;  gtidx[i1] = bj1;

    if (lab0 == 1) atomicAdd(&sm.cnt[0], 1);
    if (lab0 == 0) atomicAdd(&sm.cnt[1], 1);
    if (lab1 == 1) atomicAdd(&sm.cnt[2], 1);
    if (lab1 == 0) atomicAdd(&sm.cnt[3], 1);
    __syncthreads();
    if (tid == 0) {
        const int b2 = 2 * blockIdx.x;
        blkpos[b2]     = sm.cnt[0];
        blkneg[b2]     = sm.cnt[1];
        blkpos[b2 + 1] = sm.cnt[2];
        blkneg[b2 + 1] = sm.cnt[3];
    }
}

// ---------------------------------------------------------------------------
// Kernel 2: exclusive scan of the 1024 per-group counts (single block).
// ---------------------------------------------------------------------------
__global__ __launch_bounds__(1024) void scan_kernel(
    const int* __restrict__ blkpos, const int* __restrict__ blkneg,
    int* __restrict__ posoff, int* __restrict__ negoff, int* __restrict__ totals)
{
    __shared__ int sp[NBLK];
    __shared__ int sn[NBLK];
    const int t = threadIdx.x;
    const int vp0 = blkpos[t];
    const int vn0 = blkneg[t];
    sp[t] = vp0; sn[t] = vn0;
    __syncthreads();
    for (int d = 1; d < NBLK; d <<= 1) {
        const int vp = (t >= d) ? sp[t - d] : 0;
        const int vn = (t >= d) ? sn[t - d] : 0;
        __syncthreads();
        sp[t] += vp; sn[t] += vn;
        __syncthreads();
    }
    posoff[t] = sp[t] - vp0;   // exclusive
    negoff[t] = sn[t] - vn0;
    if (t == NBLK - 1) { totals[0] = sp[t]; totals[1] = sn[t]; }
}

// ---------------------------------------------------------------------------
// Kernel 3: emit the first MAX_POS positive / TOTAL negative anchor indices in
// anchor order, using wave32 ballot ranks (deterministic, no global atomics).
// ---------------------------------------------------------------------------
__global__ __launch_bounds__(256) void emit_kernel(
    const int* __restrict__ labels,
    const int* __restrict__ posoff, const int* __restrict__ negoff,
    int* __restrict__ pos_list, int* __restrict__ neg_list)
{
    __shared__ int wpos[8];
    __shared__ int wneg[8];
    const int tid  = threadIdx.x;
    const int i    = blockIdx.x * 256 + tid;
    const int lab  = labels[i];
#if __has_builtin(__builtin_amdgcn_ballot_w32)
    const unsigned mp = __builtin_amdgcn_ballot_w32(lab == 1);
    const unsigned mn = __builtin_amdgcn_ballot_w32(lab == 0);
#else
    const unsigned mp = (unsigned)__ballot(lab == 1);
    const unsigned mn = (unsigned)__ballot(lab == 0);
#endif
    const int lane = tid & 31;
    const int wid  = tid >> 5;
    if (lane == 0) { wpos[wid] = __popc(mp); wneg[wid] = __popc(mn); }
    __syncthreads();
    int bp = 0, bn = 0;
    for (int w = 0; w < 8; ++w) {
        if (w < wid) { bp += wpos[w]; bn += wneg[w]; }
    }
    const unsigned below = (1u << lane) - 1u;
    if (lab == 1) {
        const int r = posoff[blockIdx.x] + bp + __popc(mp & below);
        if (r < MAX_POS) pos_list[r] = i;
    } else if (lab == 0) {
        const int r = negoff[blockIdx.x] + bn + __popc(mn & below);
        if (r < TOTAL) neg_list[r] = i;
    }
}

// ---------------------------------------------------------------------------
// Kernel 4: assemble out_idx / sel_labels / sel_gt / cur_pos.
// ---------------------------------------------------------------------------
__global__ __launch_bounds__(256) void finalize_kernel(
    const int* __restrict__ labels, const int* __restrict__ gtidx,
    const int* __restrict__ totals,
    const int* __restrict__ pos_list, const int* __restrict__ neg_list,
    int* __restrict__ out)
{
    const int t = threadIdx.x;
    const int total_pos = totals[0];
    const int total_neg = totals[1];
    const int cur_pos = imin(total_pos, MAX_POS);
    const int cur_neg = imin(TOTAL - cur_pos, total_neg);

    int idx = -1, lab = -1, g = -1;
    if (t < cur_pos) {
        idx = pos_list[t];
    } else if (t < cur_pos + cur_neg) {
        idx = neg_list[t - cur_pos];
    }
    if (idx >= 0) { lab = labels[idx]; g = gtidx[idx]; }
    out[t]       = idx;
    out[256 + t] = lab;
    out[512 + t] = g;
    if (t == 0) out[768] = cur_pos;
}

// ---------------------------------------------------------------------------
extern "C" void kernel_launch(void* const* d_in, const int* in_sizes, int n_in,
                              void* d_out, int out_size, void* d_ws, size_t ws_size,
                              hipStream_t stream)
{
    (void)in_sizes; (void)n_in; (void)out_size; (void)ws_size;

    const float4* anchors = (const float4*)d_in[0];
    const float*  gtb     = (const float*)d_in[1];
    const int*    p_img_h = (const int*)d_in[2];
    const int*    p_img_w = (const int*)d_in[3];

    char* ws = (char*)d_ws;
    int* labels   = (int*)(ws);
    int* gtidx    = (int*)(ws + (size_t)N_ANCH * 4);
    int* blkpos   = (int*)(ws + (size_t)N_ANCH * 8);
    int* blkneg   = blkpos + NBLK;
    int* posoff   = blkneg + NBLK;
    int* negoff   = posoff + NBLK;
    int* totals   = negoff + NBLK;
    int* pos_list = totals + 2;
    int* neg_list = pos_list + MAX_POS;
    int* out      = (int*)d_out;

    iou_label_kernel<<<NBLK_A, 256, 0, stream>>>(anchors, gtb, p_img_h, p_img_w,
                                                 labels, gtidx, blkpos, blkneg);
    scan_kernel<<<1, NBLK, 0, stream>>>(blkpos, blkneg, posoff, negoff, totals);
    emit_kernel<<<NBLK, 256, 0, stream>>>(labels, posoff, negoff, pos_list, neg_list);
    finalize_kernel<<<1, 256, 0, stream>>>(labels, gtidx, totals, pos_list, neg_list, out);
}